// DOMINANT_MODEL_73194832659151
// MI455X (gfx1250) — compile-verified
//
#include <hip/hip_runtime.h>

typedef float v2f __attribute__((ext_vector_type(2)));
typedef float v8f __attribute__((ext_vector_type(8)));

#define NODES 16384
#define NIN   256
#define NHID  64

// ---------------------------------------------------------------------------
// fp32 WMMA GEMM, fully compile-time shaped:
//   C[NODES, NOUT] = A[NODES, K] @ B
//   BT=false: B is row-major [K, NOUT]        (weight matrices)
//   BT=true : B[k][n] = Bsrc[n*K + k]         (z.T for the structure decoder)
// One wave computes CT consecutive 16x16 column tiles of one 16-row strip,
// keeping its A fragment (K/4 x v2f) resident across tiles.
// 8 waves / 256-thread block. Grids are sized exactly (all dims divide).
// ---------------------------------------------------------------------------
template <int K, int NOUT, bool BT, int CT>
__global__ __launch_bounds__(256)
void gemm_wmma(const float* __restrict__ A, const float* __restrict__ B,
               float* __restrict__ C) {
    constexpr int GRPS_PER_ROW = NOUT / (16 * CT);   // power of two
    const int lane = threadIdx.x & 31;
    const int wave = threadIdx.x >> 5;
    const unsigned w = blockIdx.x * 8u + wave;
    const int tileRow  = (int)(w / GRPS_PER_ROW);
    const int tileGrp  = (int)(w % GRPS_PER_ROW);
    const int rowBase  = tileRow * 16;
    const int colBase0 = tileGrp * (16 * CT);

    const int mn = lane & 15;                 // A row within tile / B col within tile
    const int kb = (lane >> 4) << 1;          // lanes 0-15 -> K 0,1 ; lanes 16-31 -> K 2,3

    // Preload A fragment: per lane, pairs (k+kb, k+kb+1) for k = 0,4,8,...  (8B aligned)
    v2f areg[K / 4];
    {
        const float* __restrict__ Arow = A + (size_t)(rowBase + mn) * K + kb;
#pragma unroll
        for (int i = 0; i < K / 4; ++i)
            areg[i] = *(const v2f*)(Arow + 4 * i);
    }

#pragma unroll
    for (int ct = 0; ct < CT; ++ct) {
        const int colBase = colBase0 + ct * 16;
        v8f acc = {};
#pragma unroll
        for (int i = 0; i < K / 4; ++i) {
            const int k = 4 * i;
            v2f b;
            if (BT) {
                // contiguous in k -> single b64 load
                b = *(const v2f*)(B + (size_t)(colBase + mn) * K + k + kb);
            } else {
                b.x = B[(size_t)(k + kb) * NOUT + colBase + mn];
                b.y = B[(size_t)(k + kb + 1) * NOUT + colBase + mn];
            }
            acc = __builtin_amdgcn_wmma_f32_16x16x4_f32(
                false, areg[i], false, b, (short)0, acc, false, false);
        }
        // C/D layout: VGPR r -> M = r + (lane>=16 ? 8 : 0), N = lane&15
        const int col  = colBase + (lane & 15);
        const int mOff = (lane >> 4) << 3;
        float* __restrict__ Crow = C + (size_t)(rowBase + mOff) * NOUT + col;
#pragma unroll
        for (int r = 0; r < 8; ++r) Crow[(size_t)r * NOUT] = acc[r];
    }
}

// ---------------------------------------------------------------------------
// Degree / normalization / aggregation / activation
// ---------------------------------------------------------------------------
__global__ void deg_init_k(float* __restrict__ deg, int n) {
    int i = blockIdx.x * blockDim.x + threadIdx.x;
    if (i < n) deg[i] = 1.0f;                 // self-loop contribution
}

__global__ void deg_accum_k(const int* __restrict__ dst, float* __restrict__ deg, int e) {
    int i = blockIdx.x * blockDim.x + threadIdx.x;
    if (i < e) atomicAdd(&deg[dst[i]], 1.0f);
}

__global__ void dinv_k(float* __restrict__ d, int n) {
    int i = blockIdx.x * blockDim.x + threadIdx.x;
    if (i < n) d[i] = rsqrtf(d[i]);           // deg >= 1 always (self loops)
}

// agg[i,f] = bias[f] + tmp[i,f] * dinv[i]^2   (bias + self-loop edge)
__global__ void init_agg_k(float* __restrict__ agg, const float* __restrict__ tmp,
                           const float* __restrict__ dinv, const float* __restrict__ bias,
                           long total, int F, int FLOG) {
    long t = (long)blockIdx.x * blockDim.x + threadIdx.x;
    if (t >= total) return;
    int i = (int)(t >> FLOG);
    int f = (int)(t & (F - 1));
    float di = dinv[i];
    agg[t] = bias[f] + tmp[t] * di * di;
}

// agg[dst,f] += tmp[src,f] * dinv[src] * dinv[dst]   for every real edge
__global__ void edge_scatter_k(float* __restrict__ agg, const float* __restrict__ tmp,
                               const float* __restrict__ dinv,
                               const int* __restrict__ src, const int* __restrict__ dst,
                               long total, int F, int FLOG) {
    long t = (long)blockIdx.x * blockDim.x + threadIdx.x;
    if (t >= total) return;
    long e = t >> FLOG;
    int f = (int)(t & (F - 1));
    int s = src[e];
    int d = dst[e];
    float v = tmp[(size_t)s * F + f] * dinv[s] * dinv[d];
    atomicAdd(&agg[(size_t)d * F + f], v);
}

__global__ void relu_k(float* __restrict__ x, long n) {
    long i = (long)blockIdx.x * blockDim.x + threadIdx.x;
    if (i < n) x[i] = fmaxf(x[i], 0.0f);
}

// ---------------------------------------------------------------------------
// Launch
// ---------------------------------------------------------------------------
extern "C" void kernel_launch(void* const* d_in, const int* in_sizes, int n_in,
                              void* d_out, int out_size, void* d_ws, size_t ws_size,
                              hipStream_t stream) {
    const float* x  = (const float*)d_in[0];
    const int*   ei = (const int*)d_in[1];
    const float* W1 = (const float*)d_in[2];
    const float* b1 = (const float*)d_in[3];
    const float* W2 = (const float*)d_in[4];
    const float* b2 = (const float*)d_in[5];
    const float* Wd = (const float*)d_in[6];
    const float* bd = (const float*)d_in[7];

    const int N = NODES;
    const int E = in_sizes[1] / 2;
    const int* src = ei;
    const int* dst = ei + E;

    float* out  = (float*)d_out;
    float* zzt  = out;                               // [N, N]
    float* xhat = out + (size_t)N * N;               // [N, NIN]

    float* dinv = (float*)d_ws;                      // [N]
    float* tmp  = dinv + N;                          // [N, NIN] (largest GEMM out)
    float* h    = tmp + (size_t)N * NIN;             // [N, NHID]
    float* z    = h   + (size_t)N * NHID;            // [N, NHID]

    const int T = 256;

    // ---- degree + symmetric norm --------------------------------------
    deg_init_k<<<(N + T - 1) / T, T, 0, stream>>>(dinv, N);
    deg_accum_k<<<(E + T - 1) / T, T, 0, stream>>>(dst, dinv, E);
    dinv_k<<<(N + T - 1) / T, T, 0, stream>>>(dinv, N);

    // ---- layer 1: h = relu(GCN(x, W1, b1)) ----------------------------
    {
        constexpr int waves = (NODES / 16) * (NHID / 16);   // 4096
        gemm_wmma<NIN, NHID, false, 1><<<waves / 8, 256, 0, stream>>>(x, W1, tmp);
        long tot = (long)N * NHID;
        init_agg_k<<<(tot + T - 1) / T, T, 0, stream>>>(h, tmp, dinv, b1, tot, NHID, 6);
        long etot = (long)E * NHID;
        edge_scatter_k<<<(etot + T - 1) / T, T, 0, stream>>>(h, tmp, dinv, src, dst, etot, NHID, 6);
        relu_k<<<(tot + T - 1) / T, T, 0, stream>>>(h, tot);
    }

    // ---- layer 2: z = relu(GCN(h, W2, b2)) ----------------------------
    {
        constexpr int waves = (NODES / 16) * (NHID / 16);   // 4096
        gemm_wmma<NHID, NHID, false, 1><<<waves / 8, 256, 0, stream>>>(h, W2, tmp);
        long tot = (long)N * NHID;
        init_agg_k<<<(tot + T - 1) / T, T, 0, stream>>>(z, tmp, dinv, b2, tot, NHID, 6);
        long etot = (long)E * NHID;
        edge_scatter_k<<<(etot + T - 1) / T, T, 0, stream>>>(z, tmp, dinv, src, dst, etot, NHID, 6);
        relu_k<<<(tot + T - 1) / T, T, 0, stream>>>(z, tot);
    }

    // ---- decoder: x_hat = relu(GCN(z, Wd, bd)) ------------------------
    {
        constexpr int waves = (NODES / 16) * (NIN / 16);    // 16384
        gemm_wmma<NHID, NIN, false, 1><<<waves / 8, 256, 0, stream>>>(z, Wd, tmp);
        long tot = (long)N * NIN;
        init_agg_k<<<(tot + T - 1) / T, T, 0, stream>>>(xhat, tmp, dinv, bd, tot, NIN, 8);
        long etot = (long)E * NIN;
        edge_scatter_k<<<(etot + T - 1) / T, T, 0, stream>>>(xhat, tmp, dinv, src, dst, etot, NIN, 8);
        relu_k<<<(tot + T - 1) / T, T, 0, stream>>>(xhat, tot);
    }

    // ---- structure decoder: zzt = z @ z.T -----------------------------
    {
        // BT=true: B[k][n] = z[n*NHID + k]; each wave does 4 column tiles
        constexpr int waves = (NODES / 16) * (NODES / (16 * 4));  // 1024 * 256
        gemm_wmma<NHID, NODES, true, 4><<<waves / 8, 256, 0, stream>>>(z, z, zzt);
    }
}